// YOLOLoss_85349590106614
// MI455X (gfx1250) — compile-verified
//
#include <hip/hip_runtime.h>
#include <cstdint>

#define BATCH 16
#define NA 19200
#define NT 50
#define NC 80
#define EPSF 1e-6f
#define LAMBDA_COORD 5.0f
#define OBJ_BLOCKS 512
#define PAIRS (BATCH * NT)

// K1 tiling: 5 target-groups of 10 targets, 6 anchor chunks of 3200
#define TGROUPS 5
#define TG 10
#define ACHUNKS 6
#define ACHUNK 3200

__device__ __forceinline__ float bce_logits(float x, float z) {
    // max(x,0) - x*z + log1p(exp(-|x|))
    return fmaxf(x, 0.0f) - x * z + log1pf(expf(-fabsf(x)));
}

// ---------------------------------------------------------------------------
// K1: partial argmax of IoU. Block = (b, target-group of 10, anchor chunk of
// 3200). Each anchor box is loaded ONCE (b128) and scored against 10 targets
// held in registers -> 5x L2 read amplification instead of 50x. Target boxes
// staged via CDNA5 async global->LDS (ASYNCcnt). Cross-lane argmax uses
// wave32 shfl_xor butterflies (no barriers); one barrier merges 8 waves.
// Tie-break everywhere: lowest anchor index (jnp.argmax semantics).
// ---------------------------------------------------------------------------
__global__ __launch_bounds__(256) void k_iou_partial(
    const float4* __restrict__ pred_boxes,
    const float*  __restrict__ target_boxes,
    float* __restrict__ part_v,
    int*   __restrict__ part_i)
{
    const int ac  = blockIdx.x % ACHUNKS;
    const int tg  = (blockIdx.x / ACHUNKS) % TGROUPS;
    const int b   = blockIdx.x / (ACHUNKS * TGROUPS);
    const int tid  = threadIdx.x;
    const int lane = tid & 31;
    const int wave = tid >> 5;

    __shared__ float s_tb[TG * 4];        // 10 target boxes
    __shared__ float s_wv[8 * TG];        // per-wave winners
    __shared__ int   s_wi[8 * TG];

    // Async DMA the 10 target boxes (40 dwords) of this group into LDS.
    const float* g = target_boxes + ((size_t)b * NT + tg * TG) * 4;
    if (tid < TG * 4) {
        asm volatile(
            "global_load_async_to_lds_b32 %0, %1, off"
            :
            : "v"((uint32_t)(uintptr_t)(&s_tb[tid])), "v"(g + tid)
            : "memory");
    }
    asm volatile("s_wait_asynccnt 0" ::: "memory");
    __syncthreads();

    float tx1[TG], ty1[TG], tx2[TG], ty2[TG], ta[TG];
    #pragma unroll
    for (int j = 0; j < TG; ++j) {
        tx1[j] = s_tb[j * 4 + 0];
        ty1[j] = s_tb[j * 4 + 1];
        tx2[j] = s_tb[j * 4 + 2];
        ty2[j] = s_tb[j * 4 + 3];
        ta[j]  = (tx2[j] - tx1[j]) * (ty2[j] - ty1[j]);
    }

    float bv[TG];
    int   bi[TG];
    #pragma unroll
    for (int j = 0; j < TG; ++j) { bv[j] = -1.0f; bi[j] = 0; }

    const int a0 = ac * ACHUNK;
    const float4* pb = pred_boxes + (size_t)b * NA + a0;
    for (int a = tid; a < ACHUNK; a += 256) {
        float4 p = pb[a];                       // x1,y1,x2,y2
        float a1 = (p.z - p.x) * (p.w - p.y);
        const int ga = a0 + a;
        #pragma unroll
        for (int j = 0; j < TG; ++j) {
            float iw    = fminf(p.z, tx2[j]) - fmaxf(p.x, tx1[j]);
            float ih    = fminf(p.w, ty2[j]) - fmaxf(p.y, ty1[j]);
            float inter = fmaxf(iw, 0.0f) * fmaxf(ih, 0.0f);
            float iou   = inter / (a1 + ta[j] - inter + EPSF);
            if (iou > bv[j]) { bv[j] = iou; bi[j] = ga; }  // strict > keeps lowest a
        }
    }

    // wave32 butterfly argmax per target, then single-barrier 8-wave merge
    #pragma unroll
    for (int j = 0; j < TG; ++j) {
        float v = bv[j];
        int   i = bi[j];
        #pragma unroll
        for (int off = 16; off > 0; off >>= 1) {
            float ov = __shfl_xor(v, off, 32);
            int   oi = __shfl_xor(i, off, 32);
            if (ov > v || (ov == v && oi < i)) { v = ov; i = oi; }
        }
        if (lane == 0) { s_wv[wave * TG + j] = v; s_wi[wave * TG + j] = i; }
    }
    __syncthreads();
    if (tid < TG) {
        float v = s_wv[tid];
        int   i = s_wi[tid];
        #pragma unroll
        for (int w = 1; w < 8; ++w) {
            float ov = s_wv[w * TG + tid];
            int   oi = s_wi[w * TG + tid];
            if (ov > v || (ov == v && oi < i)) { v = ov; i = oi; }
        }
        const int t = tg * TG + tid;
        const int p = (b * NT + t) * ACHUNKS + ac;
        part_v[p] = v;
        part_i[p] = i;
    }
}

// ---------------------------------------------------------------------------
// K1b: merge the ACHUNKS partials per (b,t) -> final best anchor index.
// ---------------------------------------------------------------------------
__global__ __launch_bounds__(256) void k_merge(
    const float* __restrict__ part_v,
    const int*   __restrict__ part_i,
    int* __restrict__ ws_best)
{
    const int p = blockIdx.x * 256 + threadIdx.x;
    if (p < PAIRS) {
        float v = part_v[p * ACHUNKS];
        int   i = part_i[p * ACHUNKS];
        for (int c = 1; c < ACHUNKS; ++c) {
            float ov = part_v[p * ACHUNKS + c];
            int   oi = part_i[p * ACHUNKS + c];
            if (ov > v || (ov == v && oi < i)) { v = ov; i = oi; }
        }
        ws_best[p] = i;
    }
}

// ---------------------------------------------------------------------------
// K2: baseline objectness BCE with z=0 over all B*A confidences.
// obj correction (-x at unique best anchors) is handled in K3.
// ---------------------------------------------------------------------------
__global__ __launch_bounds__(256) void k_obj_base(
    const float* __restrict__ pred_conf,
    float* __restrict__ ws_obj)
{
    __shared__ float s_red[256];
    const int tid = threadIdx.x;
    float acc = 0.0f;
    for (int i = blockIdx.x * 256 + tid; i < BATCH * NA; i += OBJ_BLOCKS * 256) {
        float x = pred_conf[i];
        acc += fmaxf(x, 0.0f) + log1pf(expf(-fabsf(x)));   // bce(x, 0)
    }
    s_red[tid] = acc;
    __syncthreads();
    for (int s = 128; s > 0; s >>= 1) {
        if (tid < s) s_red[tid] += s_red[tid + s];
        __syncthreads();
    }
    if (tid == 0) ws_obj[blockIdx.x] = s_red[0];
}

// ---------------------------------------------------------------------------
// K3: per (b,t) pair — class BCE sum (80 logits), CIoU loss, and obj-mask
// correction (-pred_conf[best], counted once per unique best anchor per image).
// ---------------------------------------------------------------------------
__global__ __launch_bounds__(128) void k_pair(
    const float4* __restrict__ pred_boxes,
    const float*  __restrict__ pred_conf,
    const float*  __restrict__ pred_cls,
    const float4* __restrict__ target_boxes,
    const int*    __restrict__ labels,
    const int*    __restrict__ ws_best,
    float* __restrict__ ws_box,
    float* __restrict__ ws_cls,
    float* __restrict__ ws_corr)
{
    __shared__ float s_red[128];
    const int tid  = threadIdx.x;
    const int pair = blockIdx.x;
    const int b = pair / NT;
    const int t = pair % NT;
    const int best = ws_best[pair];
    const int lab  = labels[pair];

    float cl = 0.0f;
    if (tid < NC) {
        float x = pred_cls[((size_t)b * NA + best) * NC + tid];
        cl = bce_logits(x, (tid == lab) ? 1.0f : 0.0f);
    }
    s_red[tid] = cl;
    __syncthreads();
    for (int s = 64; s > 0; s >>= 1) {
        if (tid < s) s_red[tid] += s_red[tid + s];
        __syncthreads();
    }

    if (tid == 0) {
        ws_cls[pair] = s_red[0];

        // obj correction: count each distinct best anchor once per image
        int unique = 1;
        for (int t2 = 0; t2 < t; ++t2)
            if (ws_best[b * NT + t2] == best) { unique = 0; break; }
        ws_corr[pair] = unique ? pred_conf[(size_t)b * NA + best] : 0.0f;

        // CIoU loss (matches reference arithmetic exactly)
        float4 p = pred_boxes[(size_t)b * NA + best];
        float4 q = target_boxes[pair];
        float area1 = (p.z - p.x) * (p.w - p.y);
        float area2 = (q.z - q.x) * (q.w - q.y);
        float iw = fmaxf(fminf(p.z, q.z) - fmaxf(p.x, q.x), 0.0f);
        float ih = fmaxf(fminf(p.w, q.w) - fmaxf(p.y, q.y), 0.0f);
        float inter = iw * ih;
        float iou = inter / (area1 + area2 - inter + EPSF);
        float cdx = fmaxf(p.z, q.z) - fminf(p.x, q.x);
        float cdy = fmaxf(p.w, q.w) - fminf(p.y, q.y);
        float c_diag = cdx * cdx + cdy * cdy;
        float ddx = (p.x + p.z - q.x - q.z) * 0.5f;
        float ddy = (p.y + p.w - q.y - q.w) * 0.5f;
        float center = ddx * ddx + ddy * ddy;
        float w1 = p.z - p.x, h1 = p.w - p.y;
        float w2 = q.z - q.x, h2 = q.w - q.y;
        const float PI = 3.14159265358979323846f;
        float dat = atanf(w2 / h2) - atanf(w1 / h1);
        float v = (4.0f / (PI * PI)) * dat * dat;
        float alpha = v / (1.0f - iou + v + EPSF);
        float ciou = iou - center / c_diag - alpha * v;
        ws_box[pair] = 1.0f - ciou;
    }
}

// ---------------------------------------------------------------------------
// K4: deterministic final reduction -> scalar loss
// ---------------------------------------------------------------------------
__global__ __launch_bounds__(256) void k_finalize(
    const float* __restrict__ ws_obj,
    const float* __restrict__ ws_box,
    const float* __restrict__ ws_cls,
    const float* __restrict__ ws_corr,
    float* __restrict__ out)
{
    __shared__ float s_red[256];
    const int tid = threadIdx.x;

    float obj = 0.0f, box = 0.0f, cls = 0.0f, corr = 0.0f;
    for (int i = tid; i < OBJ_BLOCKS; i += 256) obj += ws_obj[i];
    for (int i = tid; i < PAIRS; i += 256) {
        box  += ws_box[i];
        cls  += ws_cls[i];
        corr += ws_corr[i];
    }

    auto block_reduce = [&](float v) -> float {
        s_red[tid] = v;
        __syncthreads();
        for (int s = 128; s > 0; s >>= 1) {
            if (tid < s) s_red[tid] += s_red[tid + s];
            __syncthreads();
        }
        float r = s_red[0];
        __syncthreads();
        return r;
    };

    float obj_s  = block_reduce(obj);
    float box_s  = block_reduce(box);
    float cls_s  = block_reduce(cls);
    float corr_s = block_reduce(corr);

    if (tid == 0) {
        float total = LAMBDA_COORD * box_s / (float)(BATCH * NT)
                    + (obj_s - corr_s) / (float)(BATCH * NA)
                    + cls_s / (float)(BATCH * NT * NC);
        out[0] = total;
    }
}

extern "C" void kernel_launch(void* const* d_in, const int* in_sizes, int n_in,
                              void* d_out, int out_size, void* d_ws, size_t ws_size,
                              hipStream_t stream) {
    const float4* pred_boxes     = (const float4*)d_in[0];
    const float*  pred_conf      = (const float*)d_in[1];
    const float*  pred_cls       = (const float*)d_in[2];
    const float*  target_boxes_f = (const float*)d_in[3];
    const float4* target_boxes4  = (const float4*)d_in[3];
    const int*    labels         = (const int*)d_in[4];
    // d_in[5] = anchors: unused by the reference computation

    // workspace layout (~53 KB)
    float* part_v  = (float*)d_ws;                        // PAIRS*ACHUNKS
    int*   part_i  = (int*)(part_v + PAIRS * ACHUNKS);    // PAIRS*ACHUNKS
    int*   ws_best = (int*)(part_i + PAIRS * ACHUNKS);    // PAIRS
    float* ws_obj  = (float*)(ws_best + PAIRS);           // OBJ_BLOCKS
    float* ws_box  = ws_obj + OBJ_BLOCKS;                 // PAIRS
    float* ws_cls  = ws_box + PAIRS;                      // PAIRS
    float* ws_corr = ws_cls + PAIRS;                      // PAIRS

    k_iou_partial<<<BATCH * TGROUPS * ACHUNKS, 256, 0, stream>>>(
        pred_boxes, target_boxes_f, part_v, part_i);
    k_merge<<<(PAIRS + 255) / 256, 256, 0, stream>>>(part_v, part_i, ws_best);
    k_obj_base<<<OBJ_BLOCKS, 256, 0, stream>>>(pred_conf, ws_obj);
    k_pair<<<PAIRS, 128, 0, stream>>>(pred_boxes, pred_conf, pred_cls,
                                      target_boxes4, labels, ws_best,
                                      ws_box, ws_cls, ws_corr);
    k_finalize<<<1, 256, 0, stream>>>(ws_obj, ws_box, ws_cls, ws_corr, (float*)d_out);
}